// scene_flow_projection_slack_16071767622174
// MI455X (gfx1250) — compile-verified
//
#include <hip/hip_runtime.h>
#include <stdint.h>

typedef float    v2f  __attribute__((ext_vector_type(2)));
typedef float    v8f  __attribute__((ext_vector_type(8)));
typedef uint32_t u32x4 __attribute__((ext_vector_type(4)));
typedef uint32_t u32x8 __attribute__((ext_vector_type(8)));

#define H_C 512
#define W_C 768
#define BLOCK 256

// lane <-> lane^16 exchange (group-of-32 swizzle, xor_mask=0x10, and_mask=0x1f)
__device__ __forceinline__ float swapx16(float v) {
  return __int_as_float(__builtin_amdgcn_ds_swizzle(__float_as_int(v), 0x401F));
}

// TDM: DMA n contiguous f32 from global into LDS at byte offset lds_byte.
// D# group0/group1 built per CDNA5 ISA 8.3/8.4; 1-row 2D tile; groups 2/3 NULL.
__device__ __forceinline__ void tdm_load_f32(const float* gptr, uint32_t n, uint32_t lds_byte) {
  uint64_t ga = (uint64_t)(uintptr_t)gptr;
  u32x4 g0;
  g0[0] = 1u;                                   // count=1, user descriptor
  g0[1] = lds_byte;                             // lds_addr
  g0[2] = (uint32_t)ga;                         // global_addr[31:0]
  g0[3] = ((uint32_t)(ga >> 32) & 0x01FFFFFFu)  // global_addr[56:32]
        | (2u << 30);                           // type=2 ("image")
  u32x8 g1;
  g1[0] = 2u << 16;                             // data_size = 4B
  g1[1] = (n & 0xFFFFu) << 16;                  // tensor_dim0[15:0]
  g1[2] = (n >> 16) | (1u << 16);               // tensor_dim0[31:16] | tensor_dim1=1
  g1[3] = (n & 0xFFFFu) << 16;                  // tile_dim0 = n
  g1[4] = 1u;                                   // tile_dim1=1, tile_dim2=0
  g1[5] = n;                                    // tensor_dim0_stride[31:0]
  g1[6] = 0u;                                   // stride0[47:32] | stride1[15:0]
  g1[7] = 0u;
  asm volatile("tensor_load_to_lds %0, %1" :: "s"(g0), "s"(g1) : "memory");
}

__global__ __launch_bounds__(BLOCK) void sfp_kernel(
    const float* __restrict__ depth1, const float* __restrict__ depth2,
    const float* __restrict__ flow12, const float* __restrict__ R1,
    const float* __restrict__ R2T,   const float* __restrict__ t1,
    const float* __restrict__ t2,    const float* __restrict__ Km,
    const float* __restrict__ Kiv,   const float* __restrict__ sflow,
    float* __restrict__ out)
{
  // LDS: [0..8]=R1 [9..17]=R2T [18..26]=K [27..35]=Kinv [36..38]=t1 [39..41]=t2
  //      [48..79]=WMMA A-matrix rows 0..7 (4 coeffs each)
  __shared__ float sm[96];
  const int tid = threadIdx.x;
  const int b   = blockIdx.y;
  // flat shared pointer: addr[31:0] is the LDS byte offset (aperture rule)
  const uint32_t smb = (uint32_t)(uintptr_t)(&sm[0]);

  if (tid == 0) {
    tdm_load_f32(R1  + b * 9, 9, smb + 0  * 4);
    tdm_load_f32(R2T + b * 9, 9, smb + 9  * 4);
    tdm_load_f32(Km  + b * 9, 9, smb + 18 * 4);
    tdm_load_f32(Kiv + b * 9, 9, smb + 27 * 4);
    tdm_load_f32(t1  + b * 3, 3, smb + 36 * 4);
    tdm_load_f32(t2  + b * 3, 3, smb + 39 * 4);
    __builtin_amdgcn_s_wait_tensorcnt(0);

    const float* r1 = sm;       // R1
    const float* rt = sm + 9;   // R2T
    const float* km = sm + 18;  // K
    const float* ki = sm + 27;  // Kinv
    float M1[9], M2[9], M3[9];
#pragma unroll
    for (int m = 0; m < 3; ++m)
#pragma unroll
      for (int j = 0; j < 3; ++j)
        M1[m*3+j] = ki[m*3+0]*r1[j] + ki[m*3+1]*r1[3+j] + ki[m*3+2]*r1[6+j];   // Kinv*R1
#pragma unroll
    for (int m = 0; m < 3; ++m)
#pragma unroll
      for (int j = 0; j < 3; ++j)
        M2[m*3+j] = M1[m*3+0]*rt[j] + M1[m*3+1]*rt[3+j] + M1[m*3+2]*rt[6+j];   // *R2T
#pragma unroll
    for (int m = 0; m < 3; ++m)
#pragma unroll
      for (int j = 0; j < 3; ++j)
        M3[m*3+j] = M2[m*3+0]*km[j] + M2[m*3+1]*km[3+j] + M2[m*3+2]*km[6+j];   // *K
    const float dt0 = sm[36]-sm[39], dt1 = sm[37]-sm[40], dt2 = sm[38]-sm[41]; // t1-t2
    float tt2[3], tt3[3];
#pragma unroll
    for (int j = 0; j < 3; ++j) tt2[j] = dt0*rt[j] + dt1*rt[3+j] + dt2*rt[6+j];
#pragma unroll
    for (int j = 0; j < 3; ++j) tt3[j] = tt2[0]*km[j] + tt2[1]*km[3+j] + tt2[2]*km[6+j];
    // A-matrix rows: 0-2 -> global_p1, 3-5 -> p1cam2_static, 6-7 -> img2_static.xy
    // A[r][k] = coeff of u_k; u = (d1*x, d1*y, d1, 1); column 3 = translation.
#pragma unroll
    for (int r = 0; r < 3; ++r) {
      sm[48 + r*4 + 0] = M1[0+r]; sm[48 + r*4 + 1] = M1[3+r];
      sm[48 + r*4 + 2] = M1[6+r]; sm[48 + r*4 + 3] = sm[36+r];
      sm[48 + (3+r)*4 + 0] = M2[0+r]; sm[48 + (3+r)*4 + 1] = M2[3+r];
      sm[48 + (3+r)*4 + 2] = M2[6+r]; sm[48 + (3+r)*4 + 3] = tt2[r];
    }
#pragma unroll
    for (int c = 0; c < 2; ++c) {
      sm[48 + (6+c)*4 + 0] = M3[0+c]; sm[48 + (6+c)*4 + 1] = M3[3+c];
      sm[48 + (6+c)*4 + 2] = M3[6+c]; sm[48 + (6+c)*4 + 3] = tt3[c];
    }
  }
  __syncthreads();

  const int p  = blockIdx.x * BLOCK + tid;
  const int xi = p % W_C;
  const int yi = p / W_C;
  const float xf = (float)xi, yf = (float)yi;
  const size_t HW  = (size_t)H_C * W_C;
  const size_t pix = (size_t)b * HW + (size_t)p;
  const size_t N   = (size_t)gridDim.y * HW;

  // Streaming inputs (read-once -> non-temporal). depth2 stays cached (gather reuse).
  const float d1  = __builtin_nontemporal_load(depth1 + pix);
  const float d2c = depth2[pix];
  const v2f   fl  = __builtin_nontemporal_load((const v2f*)flow12 + pix);
  const float sf0 = __builtin_nontemporal_load(sflow + pix*3 + 0);
  const float sf1 = __builtin_nontemporal_load(sflow + pix*3 + 1);
  const float sf2 = __builtin_nontemporal_load(sflow + pix*3 + 2);

  // ---- batched affine transforms via V_WMMA_F32_16X16X4_F32 ----
  // A 16x4 layout: lanes 0-15 hold row (lane&15) comps (0,1); lanes 16-31 comps (2,3).
  // Rows 8-15 duplicate rows 0-7 so WMMA#1 serves pixels 0-15 (D rows 0-7, lanes 0-15)
  // and WMMA#2 serves pixels 16-31 (D rows 8-15, lanes 16-31) with no output permute.
  const int  lane = tid & 31;
  const bool lo   = (lane & 16) == 0;
  const int  ar   = (lane & 7) * 4;
  v2f A;
  A.x = sm[48 + ar + (lo ? 0 : 2)];
  A.y = sm[48 + ar + (lo ? 1 : 3)];

  const float u0 = d1 * xf, u1 = d1 * yf, u2 = d1, u3 = 1.0f;
  const float u0s = swapx16(u0), u1s = swapx16(u1), u2s = swapx16(u2), u3s = swapx16(u3);
  // B 4x16 layout: lanes 0-15 hold K=0,1 (v0,v1); lanes 16-31 hold K=2,3.
  v2f B1, B2;
  B1.x = lo ? u0  : u2s;  B1.y = lo ? u1  : u3s;   // columns = pixels 0-15
  B2.x = lo ? u0s : u2;   B2.y = lo ? u1s : u3;    // columns = pixels 16-31
  v8f c0 = {0.f,0.f,0.f,0.f,0.f,0.f,0.f,0.f};
  v8f D1 = __builtin_amdgcn_wmma_f32_16x16x4_f32(false, A, false, B1, (short)0, c0, false, false);
  v8f D2 = __builtin_amdgcn_wmma_f32_16x16x4_f32(false, A, false, B2, (short)0, c0, false, false);

  const float g1x = lo ? D1[0] : D2[0];  // global_p1
  const float g1y = lo ? D1[1] : D2[1];
  const float g1z = lo ? D1[2] : D2[2];
  const float ps0 = lo ? D1[3] : D2[3];  // p1_camera_2_static
  const float ps1 = lo ? D1[4] : D2[4];
  const float ps2 = lo ? D1[5] : D2[5];
  const float isx = lo ? D1[6] : D2[6];  // p1_image_2_static.xy
  const float isy = lo ? D1[7] : D2[7];
  const float isz = ps0*sm[20] + ps1*sm[23] + ps2*sm[26];  // p1c2s @ K[:,2]

  // dynamic scene-flow part: sflow@R2T, then through K
  const float sr0 = sf0*sm[9]  + sf1*sm[12] + sf2*sm[15];
  const float sr1 = sf0*sm[10] + sf1*sm[13] + sf2*sm[16];
  const float sr2 = sf0*sm[11] + sf1*sm[14] + sf2*sm[17];
  const float pc0 = ps0 + sr0, pc1 = ps1 + sr1, pc2 = ps2 + sr2;  // p1_camera_2
  const float i2x = pc0*sm[18] + pc1*sm[21] + pc2*sm[24];
  const float i2y = pc0*sm[19] + pc1*sm[22] + pc2*sm[25];
  const float i2z = pc0*sm[20] + pc1*sm[23] + pc2*sm[26];

  // projections (bad -> coord -> zero flow)
  const float invz  = 1.0f / (i2z + 1e-8f);
  const bool  ok    = !(i2z < 0.001f);
  const float dfx   = ok ? (i2x*invz - xf) : 0.0f;
  const float dfy   = ok ? (i2y*invz - yf) : 0.0f;
  const float invzs = 1.0f / (isz + 1e-8f);
  const bool  oks   = !(isz < 0.001f);
  const float dsx   = oks ? (isx*invzs - xf) : 0.0f;
  const float dsy   = oks ? (isy*invzs - yf) : 0.0f;

  // backward warp (bilinear, border). warped_p2cam2 = (Sum w*d2*[xc,yc,1]) @ Kinv
  const float gx = xf + fl.x, gy = yf + fl.y;
  const float x0 = floorf(gx), y0 = floorf(gy);
  const float wx = gx - x0,   wy = gy - y0;
  const float xc0 = fminf(fmaxf(x0,        0.0f), (float)(W_C-1));
  const float xc1 = fminf(fmaxf(x0 + 1.0f, 0.0f), (float)(W_C-1));
  const float yc0 = fminf(fmaxf(y0,        0.0f), (float)(H_C-1));
  const float yc1 = fminf(fmaxf(y0 + 1.0f, 0.0f), (float)(H_C-1));
  const int x0i = (int)xc0, x1i = (int)xc1, y0i = (int)yc0, y1i = (int)yc1;
  const float* d2b = depth2 + (size_t)b * HW;
  const float v00 = d2b[y0i*W_C + x0i];
  const float v01 = d2b[y0i*W_C + x1i];
  const float v10 = d2b[y1i*W_C + x0i];
  const float v11 = d2b[y1i*W_C + x1i];
  const float a00 = (1.0f-wx)*(1.0f-wy)*v00;
  const float a01 = wx*(1.0f-wy)*v01;
  const float a10 = (1.0f-wx)*wy*v10;
  const float a11 = wx*wy*v11;
  const float dw  = a00 + a01 + a10 + a11;               // depth_warp_1_2
  const float qx  = (a00 + a10)*xc0 + (a01 + a11)*xc1;
  const float qy  = (a00 + a01)*yc0 + (a10 + a11)*yc1;
  const float wp0 = qx*sm[27] + qy*sm[30] + dw*sm[33];
  const float wp1 = qx*sm[28] + qy*sm[31] + dw*sm[34];
  const float wp2 = qx*sm[29] + qy*sm[32] + dw*sm[35];

  // ---- outputs (write-once -> non-temporal), tuple order, flat-concatenated ----
  const size_t p3 = pix * 3, p2 = pix * 2;
  v2f df  = {dfx, dfy};
  v2f dfs = {dsx, dsy};
  __builtin_nontemporal_store(df,  (v2f*)(out + p2));                 // depth_flow_1_2
  __builtin_nontemporal_store(i2z, out + 2*N + pix);                  // depth_image_1_2
  __builtin_nontemporal_store(dw,  out + 3*N + pix);                  // depth_warp_1_2
  __builtin_nontemporal_store(d1,  out + 4*N + pix);                  // depth_1
  __builtin_nontemporal_store(d2c, out + 5*N + pix);                  // depth_2
  __builtin_nontemporal_store(sf0, out + 6*N + p3 + 0);               // sflow_1_2
  __builtin_nontemporal_store(sf1, out + 6*N + p3 + 1);
  __builtin_nontemporal_store(sf2, out + 6*N + p3 + 2);
  __builtin_nontemporal_store(g1x, out + 9*N + p3 + 0);               // global_p1
  __builtin_nontemporal_store(g1y, out + 9*N + p3 + 1);
  __builtin_nontemporal_store(g1z, out + 9*N + p3 + 2);
  __builtin_nontemporal_store(dfs, (v2f*)(out + 12*N + p2));          // depth_flow_1_2_static
  __builtin_nontemporal_store(pc0, out + 14*N + p3 + 0);              // p1_camera_2
  __builtin_nontemporal_store(pc1, out + 14*N + p3 + 1);
  __builtin_nontemporal_store(pc2, out + 14*N + p3 + 2);
  __builtin_nontemporal_store(wp0, out + 17*N + p3 + 0);              // warped_p2_camera_2
  __builtin_nontemporal_store(wp1, out + 17*N + p3 + 1);
  __builtin_nontemporal_store(wp2, out + 17*N + p3 + 2);
}

extern "C" void kernel_launch(void* const* d_in, const int* in_sizes, int n_in,
                              void* d_out, int out_size, void* d_ws, size_t ws_size,
                              hipStream_t stream) {
  (void)n_in; (void)out_size; (void)d_ws; (void)ws_size;
  const float* depth1 = (const float*)d_in[0];
  const float* depth2 = (const float*)d_in[1];
  const float* flow12 = (const float*)d_in[2];
  const float* R1     = (const float*)d_in[4];
  const float* R2T    = (const float*)d_in[7];
  const float* t1     = (const float*)d_in[8];
  const float* t2     = (const float*)d_in[9];
  const float* Km     = (const float*)d_in[10];
  const float* Kinv   = (const float*)d_in[11];
  const float* sflow  = (const float*)d_in[12];
  const int B = in_sizes[4] / 9;  // R_1 is (B,1,1,3,3)
  dim3 grid((H_C * W_C) / BLOCK, B);
  sfp_kernel<<<grid, BLOCK, 0, stream>>>(depth1, depth2, flow12, R1, R2T, t1, t2,
                                         Km, Kinv, sflow, (float*)d_out);
}